// PointConvUpdate_34291018891265
// MI455X (gfx1250) — compile-verified
//
#include <hip/hip_runtime.h>
#include <hip/hip_bf16.h>

// ---------------------------------------------------------------------------
// Equivariant point-conv update for MI455X (gfx1250, wave32).
//   msg  = scatter_add(edge_message, edge_dst) * 0.25           (memory bound)
//   out  = [lin + sc] per irrep block, both expressed as GEMMs and executed
//          with V_WMMA_F32_16X16X4_F32 (fp32 matrix cores, exact numerics).
// ---------------------------------------------------------------------------

#define DIM        480
#define MUL_ATTR   16

typedef __attribute__((ext_vector_type(2))) float v2f;
typedef __attribute__((ext_vector_type(8))) float v8f;

// ---------------------------------------------------------------- zero scratch
__global__ __launch_bounds__(256) void zero_f32(float* __restrict__ p, int n) {
    int i = blockIdx.x * blockDim.x + threadIdx.x;
    if (i < n) p[i] = 0.0f;
}

// ------------------------------------------------------------- scatter-add msg
// One thread per (edge, 16-byte chunk). 480 floats = 120 float4 per edge.
__global__ __launch_bounds__(256) void scatter_add_edges(
    const float* __restrict__ edge_message,   // [E,480]
    const int*   __restrict__ edge_index,     // [2,E]; dst = row 1
    float*       __restrict__ msg,            // [N,480]
    int n_edges)
{
    long long tid = (long long)blockIdx.x * blockDim.x + threadIdx.x;
    int e = (int)(tid / 120);
    int q = (int)(tid % 120);
    if (e >= n_edges) return;

    const float4 v = *(const float4*)(edge_message + (long long)e * DIM + q * 4);
    int d = edge_index[n_edges + e];          // edge_dst
    float* p = msg + (long long)d * DIM + q * 4;
    atomicAdd(p + 0, v.x);
    atomicAdd(p + 1, v.y);
    atomicAdd(p + 2, v.z);
    atomicAdd(p + 3, v.w);
}

// ----------------------------------------------------------- per-block GEMMs
// Rows r = n*C + c (C = 2l+1).  out[r, w] =
//   sum_u  s1 * msg[n, OFF + u*C + c]                      * W2 [u, w]
// + sum_uv s2 * feats[n, OFF + u*C + c] * attrs[n, v]      * Wsc[u, v, w]
// Both phases are K-accumulated into one fp32 WMMA accumulator.
// Workgroup: 128 threads (4 waves). WG tile = 64 rows x 16 cols; one 16x16
// output tile per wave.  K-chunk = 32, 8 wmma steps per chunk.
template<int MUL, int C, int OFF>
__global__ __launch_bounds__(128) void irrep_update(
    const float* __restrict__ msg,     // [N,480] (already scatter-summed)
    const float* __restrict__ feats,   // [N,480]
    const float* __restrict__ attrs,   // [N,16]
    const float* __restrict__ w2,      // [MUL,MUL]
    const float* __restrict__ wsc,     // [MUL,16,MUL] -> [MUL*16, MUL]
    float* __restrict__ out,           // [N,480]
    int n_nodes)
{
    constexpr int BM = 64;   // rows per workgroup
    constexpr int KT = 32;   // K chunk

    const int R    = n_nodes * C;
    const int col0 = blockIdx.x * 16;     // output column tile
    const int row0 = blockIdx.y * BM;     // output row tile

    __shared__ float As[BM][KT + 1];      // +1 pad: conflict-free column reads
    __shared__ float Bs[KT][16 + 1];

    const int t    = threadIdx.x;
    const int lane = t & 31;
    const int wave = t >> 5;

    const float s1 = 0.25f / __builtin_sqrtf((float)MUL);        // 1/sqrt(16)*1/sqrt(mul)
    const float s2 = 1.0f  / __builtin_sqrtf((float)(MUL * MUL_ATTR));

    v8f acc = {};

    // WMMA f32 16x16x4 fragment indices (ISA 7.12.2):
    //  A: vgpr0 = A[m][k0 + koff], vgpr1 = A[m][k0 + koff + 1],
    //     m = lane&15, koff = 0 for lanes 0-15, 2 for lanes 16-31.
    //  B mirrors A with N in place of M.
    const int fm   = lane & 15;
    const int koff = (lane >> 4) << 1;

    for (int phase = 0; phase < 2; ++phase) {
        const int K = (phase == 0) ? MUL : MUL * MUL_ATTR;
        for (int k0 = 0; k0 < K; k0 += KT) {
            // ---- stage A tile: 64 x 32, coalesced along k ----
            #pragma unroll
            for (int i = t; i < BM * KT; i += 128) {
                int r = i >> 5, k = i & 31;
                int g = row0 + r;
                float a = 0.0f;
                if (g < R) {
                    int n = g / C, c = g - n * C;
                    int kg = k0 + k;
                    if (phase == 0) {
                        a = s1 * msg[(long long)n * DIM + OFF + kg * C + c];
                    } else {
                        int u = kg >> 4, v = kg & 15;
                        a = s2 * feats[(long long)n * DIM + OFF + u * C + c]
                               * attrs[n * MUL_ATTR + v];
                    }
                }
                As[r][k] = a;
            }
            // ---- stage B tile: 32 x 16 ----
            #pragma unroll
            for (int i = t; i < KT * 16; i += 128) {
                int k = i >> 4, w = i & 15;
                int kg = k0 + k;
                Bs[k][w] = (phase == 0) ? w2 [kg * MUL + col0 + w]
                                        : wsc[kg * MUL + col0 + w];
            }
            __syncthreads();

            // ---- 8 wmma steps over this K chunk ----
            #pragma unroll
            for (int ks = 0; ks < KT; ks += 4) {
                v2f a, b;
                a.x = As[wave * 16 + fm][ks + koff];
                a.y = As[wave * 16 + fm][ks + koff + 1];
                b.x = Bs[ks + koff][fm];
                b.y = Bs[ks + koff + 1][fm];
                acc = __builtin_amdgcn_wmma_f32_16x16x4_f32(
                    /*neg_a=*/false, a, /*neg_b=*/false, b,
                    /*c_mod=*/(short)0, acc,
                    /*reuse_a=*/false, /*reuse_b=*/false);
            }
            __syncthreads();
        }
    }

    // ---- store: C/D layout — vgpr r: lanes 0-15 -> M=r, lanes 16-31 -> M=r+8
    #pragma unroll
    for (int r = 0; r < 8; ++r) {
        int mrow = (lane < 16) ? r : (r + 8);
        int g = row0 + wave * 16 + mrow;
        if (g < R) {
            int n = g / C, c = g - n * C;
            out[(long long)n * DIM + OFF + (col0 + fm) * C + c] = acc[r];
        }
    }
}

// ---------------------------------------------------------------------------
extern "C" void kernel_launch(void* const* d_in, const int* in_sizes, int n_in,
                              void* d_out, int out_size, void* d_ws, size_t ws_size,
                              hipStream_t stream)
{
    const float* node_feats   = (const float*)d_in[0];   // [N,480]
    const float* node_attrs   = (const float*)d_in[1];   // [N,16]
    const float* edge_message = (const float*)d_in[2];   // [E,480]
    const int*   edge_index   = (const int*)  d_in[3];   // [2,E]
    const float* W2_0  = (const float*)d_in[4];
    const float* W2_1  = (const float*)d_in[5];
    const float* W2_2  = (const float*)d_in[6];
    const float* Wsc_0 = (const float*)d_in[7];
    const float* Wsc_1 = (const float*)d_in[8];
    const float* Wsc_2 = (const float*)d_in[9];

    float* out = (float*)d_out;
    float* msg = (float*)d_ws;                 // [N,480] scratch

    const int n_nodes = in_sizes[0] / DIM;
    const int n_edges = in_sizes[2] / DIM;

    // 1) zero msg scratch
    {
        int n = n_nodes * DIM;
        zero_f32<<<(n + 255) / 256, 256, 0, stream>>>(msg, n);
    }
    // 2) scatter-add edge messages (memory-bound: ~307 MB streamed)
    {
        long long total = (long long)n_edges * 120;
        int blocks = (int)((total + 255) / 256);
        scatter_add_edges<<<blocks, 256, 0, stream>>>(edge_message, edge_index,
                                                      msg, n_edges);
    }
    // 3) per-irrep fused lin + self-connection GEMMs (fp32 WMMA)
    {
        // block 0: mul=128, l=0 (C=1), offset 0
        dim3 g0(128 / 16, (n_nodes * 1 + 63) / 64);
        irrep_update<128, 1, 0><<<g0, 128, 0, stream>>>(
            msg, node_feats, node_attrs, W2_0, Wsc_0, out, n_nodes);

        // block 1: mul=64, l=1 (C=3), offset 128
        dim3 g1(64 / 16, (n_nodes * 3 + 63) / 64);
        irrep_update<64, 3, 128><<<g1, 128, 0, stream>>>(
            msg, node_feats, node_attrs, W2_1, Wsc_1, out, n_nodes);

        // block 2: mul=32, l=2 (C=5), offset 320
        dim3 g2(32 / 16, (n_nodes * 5 + 63) / 64);
        irrep_update<32, 5, 320><<<g2, 128, 0, stream>>>(
            msg, node_feats, node_attrs, W2_2, Wsc_2, out, n_nodes);
    }
}